// LogicLayer_20847771255191
// MI455X (gfx1250) — compile-verified
//
#include <hip/hip_runtime.h>
#include <cstdint>

// Problem constants (from reference setup_inputs)
constexpr int B_DIM   = 2048;
constexpr int IN_DIM  = 8192;
constexpr int OUT_DIM = 16384;

constexpr int THREADS        = 256;
constexpr int ROWS_PER_BLOCK = 2;          // 2 x 32KB rows = 64KB static LDS
constexpr int LDS_FLOATS     = ROWS_PER_BLOCK * IN_DIM;
constexpr int VEC            = 4;          // outputs per thread per iteration

typedef int   v4i __attribute__((ext_vector_type(4)));
typedef float v4f __attribute__((ext_vector_type(4)));
typedef __attribute__((address_space(1))) v4i* gptr128;   // global AS
typedef __attribute__((address_space(3))) v4i* lptr128;   // LDS AS

#if __has_builtin(__builtin_amdgcn_global_load_async_to_lds_b128)
#define HAVE_ASYNC_LDS 1
#else
#define HAVE_ASYNC_LDS 0
#endif

// Fold softmax(w[o,:]) over the 16 gates into out = C0 + C1*a + C2*b + C3*a*b.
// Gate g -> (const, a, b, ab) coefficients, difflogic order.
__device__ inline float4 fold_softmax16(const float* __restrict__ wrow) {
    float v[16];
    float m = -INFINITY;
#pragma unroll
    for (int i = 0; i < 16; ++i) { v[i] = wrow[i]; m = fmaxf(m, v[i]); }
    float s = 0.0f;
#pragma unroll
    for (int i = 0; i < 16; ++i) { v[i] = expf(v[i] - m); s += v[i]; }
    const float inv = 1.0f / s;
#pragma unroll
    for (int i = 0; i < 16; ++i) v[i] *= inv;

    const float C0 = v[8] + v[9] + v[10] + v[11] + v[12] + v[13] + v[14] + v[15];
    const float C1 = v[2] + v[3] + v[6] + v[7] - v[8] - v[9] - v[12] - v[13];
    const float C2 = v[4] + v[5] + v[6] + v[7] - v[8] - v[9] - v[10] - v[11];
    const float C3 = v[1] - v[2] - v[4] - 2.0f * v[6] - v[7]
                   + v[8] + 2.0f * v[9] + v[11] + v[13] - v[14];
    return make_float4(C0, C1, C2, C3);
}

__global__ void softmax_fold_kernel(const float* __restrict__ w,
                                    float4* __restrict__ coeffs) {
    const int o = blockIdx.x * blockDim.x + threadIdx.x;
    if (o >= OUT_DIM) return;
    coeffs[o] = fold_softmax16(w + (size_t)o * 16);
}

__global__ __launch_bounds__(THREADS)
void logic_gather_kernel(const float*  __restrict__ x,
                         const int*    __restrict__ idx_a,
                         const int*    __restrict__ idx_b,
                         const float4* __restrict__ coeffs,
                         const float*  __restrict__ weights,
                         int use_ws,
                         float* __restrict__ out) {
    __shared__ float lds[LDS_FLOATS];

    const int tid  = threadIdx.x;
    const int row0 = blockIdx.x * ROWS_PER_BLOCK;
    const float* __restrict__ xrows = x + (size_t)row0 * IN_DIM;

    // ---- Stage ROWS_PER_BLOCK contiguous x rows into LDS (async DMA path) ----
#if HAVE_ASYNC_LDS
    {
        char* gbase = (char*)xrows;
        char* lbase = (char*)lds;
        constexpr int CHUNKS = LDS_FLOATS / (THREADS * 4);   // b128 per lane
#pragma unroll
        for (int c = 0; c < CHUNKS; ++c) {
            const int byteoff = (c * THREADS + tid) * 16;
            __builtin_amdgcn_global_load_async_to_lds_b128(
                (gptr128)(gbase + byteoff),
                (lptr128)(lbase + byteoff),
                0, 0);
        }
#if __has_builtin(__builtin_amdgcn_s_wait_asynccnt)
        __builtin_amdgcn_s_wait_asynccnt(0);
#else
        asm volatile("s_wait_asynccnt 0" ::: "memory");
#endif
    }
#else
    {
        const float4* __restrict__ src = (const float4*)xrows;
        float4* dst = (float4*)lds;
        constexpr int CHUNKS = LDS_FLOATS / (THREADS * 4);
#pragma unroll
        for (int c = 0; c < CHUNKS; ++c)
            dst[c * THREADS + tid] = src[c * THREADS + tid];
    }
#endif
    __syncthreads();

    // ---- Stream output columns, 4 per thread per iteration ----
    const size_t outbase = (size_t)row0 * OUT_DIM;
    constexpr int ITERS = OUT_DIM / (THREADS * VEC);   // 16
    for (int j = 0; j < ITERS; ++j) {
        const int t  = j * THREADS + tid;
        const int o0 = t * VEC;
        const v4i ia4 = ((const v4i*)idx_a)[t];
        const v4i ib4 = ((const v4i*)idx_b)[t];

        float4 c[VEC];
        if (use_ws) {
#pragma unroll
            for (int k = 0; k < VEC; ++k) c[k] = coeffs[o0 + k];
        } else {
#pragma unroll
            for (int k = 0; k < VEC; ++k)
                c[k] = fold_softmax16(weights + (size_t)(o0 + k) * 16);
        }

#pragma unroll
        for (int rr = 0; rr < ROWS_PER_BLOCK; ++rr) {
            v4f r;
#pragma unroll
            for (int k = 0; k < VEC; ++k) {
                const float a = lds[rr * IN_DIM + ia4[k]];
                const float b = lds[rr * IN_DIM + ib4[k]];
                float acc = fmaf(c[k].w, a * b, c[k].x);   // C3*ab + C0
                acc = fmaf(c[k].y, a, acc);                // + C1*a
                acc = fmaf(c[k].z, b, acc);                // + C2*b
                r[k] = acc;
            }
            // write-once 128 MB stream: non-temporal b128 store, keep L2 for x/idx/coeffs
            __builtin_nontemporal_store(
                r, (v4f*)(out + outbase + (size_t)rr * OUT_DIM + o0));
        }
    }
}

extern "C" void kernel_launch(void* const* d_in, const int* in_sizes, int n_in,
                              void* d_out, int out_size, void* d_ws, size_t ws_size,
                              hipStream_t stream) {
    const float* x       = (const float*)d_in[0];
    const float* weights = (const float*)d_in[1];
    const int*   idx_a   = (const int*)d_in[2];
    const int*   idx_b   = (const int*)d_in[3];
    float*       out     = (float*)d_out;

    const size_t coeff_bytes = (size_t)OUT_DIM * sizeof(float4);
    const int use_ws = (ws_size >= coeff_bytes) ? 1 : 0;
    float4* coeffs = (float4*)d_ws;

    if (use_ws) {
        softmax_fold_kernel<<<OUT_DIM / THREADS, THREADS, 0, stream>>>(weights, coeffs);
    }
    logic_gather_kernel<<<B_DIM / ROWS_PER_BLOCK, THREADS, 0, stream>>>(
        x, idx_a, idx_b, coeffs, weights, use_ws, out);
}